// CentroidLoss_50182397887064
// MI455X (gfx1250) — compile-verified
//
#include <hip/hip_runtime.h>
#include <math.h>

#define BB 8192
#define DD 128
#define KK 256

typedef __attribute__((ext_vector_type(2))) float v2f;
typedef __attribute__((ext_vector_type(8))) float v8f;

// ---------------------------------------------------------------------------
// Kernel 0: precompute row norms ||h_b||^2, ||c_k||^2, zero the output scalar.
// ---------------------------------------------------------------------------
__global__ void centroid_pre(const float* __restrict__ h,
                             const float* __restrict__ centers,
                             float* __restrict__ cn,   // [KK]
                             float* __restrict__ hn,   // [BB]
                             float* __restrict__ out) {
    int i = blockIdx.x * blockDim.x + threadIdx.x;
    if (i == 0) out[0] = 0.0f;
    if (i < KK) {
        const float4* p = (const float4*)(centers + (size_t)i * DD);
        float s = 0.0f;
        #pragma unroll 8
        for (int t = 0; t < DD / 4; ++t) {
            float4 v = p[t];
            s += v.x * v.x + v.y * v.y + v.z * v.z + v.w * v.w;
        }
        cn[i] = s;
    } else if (i < KK + BB) {
        int r = i - KK;
        const float4* p = (const float4*)(h + (size_t)r * DD);
        float s = 0.0f;
        #pragma unroll 8
        for (int t = 0; t < DD / 4; ++t) {
            float4 v = p[t];
            s += v.x * v.x + v.y * v.y + v.z * v.z + v.w * v.w;
        }
        hn[r] = s;
    }
}

// ---------------------------------------------------------------------------
// Kernel 1: main loss. Each block: 16 rows of h x all 256 centers.
// 8 waves; each wave computes two 16x16 tiles of S = h @ centers^T with
// v_wmma_f32_16x16x4_f32 chained over D=128 (32 K-steps), then
// norm_sq = ||h||^2 + ||c||^2 - 2S into LDS, then per-row logsumexp + gather.
// ---------------------------------------------------------------------------
__global__ void centroid_main(const float* __restrict__ h,
                              const float* __restrict__ centers,
                              const int* __restrict__ y,
                              const float* __restrict__ cn,
                              const float* __restrict__ hn,
                              float* __restrict__ out) {
    __shared__ float nsq[16][KK + 1];   // padded stride
    __shared__ float redmax[16][16];
    __shared__ float redsum[16][16];

    const int lane = threadIdx.x & 31;
    const int w    = threadIdx.x >> 5;          // wave 0..7
    const int m16  = lane & 15;
    const int khalf = (lane >> 4) * 2;          // 0 or 2
    const int row_base = blockIdx.x * 16;

    const int colA = 32 * w + m16;              // tile 0 column for this lane
    const int colB = colA + 16;                 // tile 1 column

    const float* arow  = h       + (size_t)(row_base + m16) * DD + khalf;
    const float* b0row = centers + (size_t)colA * DD + khalf;
    const float* b1row = centers + (size_t)colB * DD + khalf;

    v8f acc0 = {0.f, 0.f, 0.f, 0.f, 0.f, 0.f, 0.f, 0.f};
    v8f acc1 = {0.f, 0.f, 0.f, 0.f, 0.f, 0.f, 0.f, 0.f};

    #pragma unroll 8
    for (int s = 0; s < DD / 4; ++s) {
        v2f a  = *(const v2f*)(arow  + 4 * s);
        v2f b0 = *(const v2f*)(b0row + 4 * s);
        v2f b1 = *(const v2f*)(b1row + 4 * s);
        acc0 = __builtin_amdgcn_wmma_f32_16x16x4_f32(false, a, false, b0,
                                                     (short)0, acc0, false, false);
        acc1 = __builtin_amdgcn_wmma_f32_16x16x4_f32(false, a, false, b1,
                                                     (short)0, acc1, false, false);
    }

    // C/D layout: VGPR r holds row M=r (lanes 0-15) or M=r+8 (lanes 16-31),
    // column N = lane&15 within the tile.
    const int rhalf = (lane >> 4) * 8;
    const float cA = cn[colA];
    const float cB = cn[colB];
    #pragma unroll
    for (int r = 0; r < 8; ++r) {
        int row_m = r + rhalf;
        float hr = hn[row_base + row_m];
        nsq[row_m][colA] = hr + cA - 2.0f * acc0[r];
        nsq[row_m][colB] = hr + cB - 2.0f * acc1[r];
    }
    __syncthreads();

    // logsumexp(-sqrt(nsq)) per row: 16 threads per row, 16 cols each.
    const int row_m = threadIdx.x >> 4;
    const int seg   = threadIdx.x & 15;
    float e[16];
    float mx = -__builtin_inff();
    #pragma unroll
    for (int i = 0; i < 16; ++i) {
        float ns = fmaxf(nsq[row_m][seg * 16 + i], 0.0f);
        float v = -sqrtf(ns);
        e[i] = v;
        mx = fmaxf(mx, v);
    }
    redmax[row_m][seg] = mx;
    __syncthreads();

    float rowmax = -__builtin_inff();
    #pragma unroll
    for (int i = 0; i < 16; ++i) rowmax = fmaxf(rowmax, redmax[row_m][i]);

    float s = 0.0f;
    #pragma unroll
    for (int i = 0; i < 16; ++i) s += expf(e[i] - rowmax);
    redsum[row_m][seg] = s;
    __syncthreads();

    if (seg == 0) {
        float tot = 0.0f;
        #pragma unroll
        for (int i = 0; i < 16; ++i) tot += redsum[row_m][i];
        float logsum = rowmax + logf(tot);
        int cls = y[row_base + row_m];
        float dist = nsq[row_m][cls];               // squared distance (reference)
        redsum[row_m][0] = dist + logsum;           // per-row contribution
    }
    __syncthreads();

    if (threadIdx.x == 0) {
        float t = 0.0f;
        #pragma unroll
        for (int rm = 0; rm < 16; ++rm) t += redsum[rm][0];
        atomicAdd(out, t * (1.0f / (float)BB));
    }
}

// ---------------------------------------------------------------------------
// Kernel 2: regularizer. Block k: pd[k][j] for all j, min over j!=k, -log, sum.
// ---------------------------------------------------------------------------
__global__ void centroid_reg(const float* __restrict__ centers,
                             const float* __restrict__ cn,
                             float* __restrict__ out) {
    __shared__ float ck[DD];
    __shared__ float red[KK];
    const int k = blockIdx.x;
    const int j = threadIdx.x;

    if (j < DD) ck[j] = centers[(size_t)k * DD + j];
    __syncthreads();

    float dot = 0.0f;
    const float* cj = centers + (size_t)j * DD;
    #pragma unroll 8
    for (int d = 0; d < DD; ++d) dot += ck[d] * cj[d];

    float pd = cn[k] + cn[j] - 2.0f * dot;
    if (j == k) pd = __builtin_inff();
    red[j] = pd;
    __syncthreads();

    for (int off = KK / 2; off > 0; off >>= 1) {
        if (j < off) red[j] = fminf(red[j], red[j + off]);
        __syncthreads();
    }
    if (j == 0) atomicAdd(out, -logf(red[0]));   // RHO = 1
}

// ---------------------------------------------------------------------------
extern "C" void kernel_launch(void* const* d_in, const int* in_sizes, int n_in,
                              void* d_out, int out_size, void* d_ws, size_t ws_size,
                              hipStream_t stream) {
    const float* h       = (const float*)d_in[0];
    const int*   y       = (const int*)d_in[1];
    const float* centers = (const float*)d_in[2];
    float* out = (float*)d_out;

    float* cn = (float*)d_ws;        // KK floats
    float* hn = cn + KK;             // BB floats

    centroid_pre<<<(KK + BB + 255) / 256, 256, 0, stream>>>(h, centers, cn, hn, out);
    centroid_main<<<BB / 16, 256, 0, stream>>>(h, centers, y, cn, hn, out);
    centroid_reg<<<KK, 256, 0, stream>>>(centers, cn, out);
}